// PartialConvInterpolation_76467597738297
// MI455X (gfx1250) — compile-verified
//
#include <hip/hip_runtime.h>
#include <hip/hip_bf16.h>

#define HH 128
#define WW 128
#define NCH 1280
#define PLANE (HH * WW)
#define LDSW 132   // padded LDS row stride: %4==0 (float4 stores), %64==4 (no bank conflicts)

typedef float v2f __attribute__((ext_vector_type(2)));
typedef float v4f __attribute__((ext_vector_type(4)));
typedef float v8f __attribute__((ext_vector_type(8)));
typedef int   v4i __attribute__((vector_size(16)));   // matches builtin param type

#define GLOBAL_AS __attribute__((address_space(1)))
#define LDS_AS    __attribute__((address_space(3)))

// ---------------------------------------------------------------------------
// Kernel 1: mask dynamics on a single [H,W] plane (mask identical per channel,
// mask_weight uniform). Produces mask_sum map of the stop iteration and T.
// ---------------------------------------------------------------------------
__global__ void __launch_bounds__(256)
mask_iter_kernel(const float* __restrict__ mask, const float* __restrict__ ref_mask,
                 const float* __restrict__ mask_weight, const int* __restrict__ max_it,
                 float* __restrict__ masksum, int* __restrict__ t_out)
{
    extern __shared__ float smem[];
    float* m  = smem;              // binary mask plane
    float* rs = smem + PLANE;      // horizontal 5-tap counts
    float* mc = smem + 2 * PLANE;  // full 5x5 counts
    __shared__ int ndiff, ncov, refcnt;

    const int tid = threadIdx.x;
    if (tid == 0) { ndiff = 0; ncov = 0; refcnt = 0; }

    int myref = 0;
    for (int v = 0; v < PLANE / 256; ++v) {
        int p = tid + 256 * v;
        m[p] = (mask[p] > 0.f) ? 1.f : 0.f;       // channel-0 plane
        if (ref_mask[p] > 0.f) myref++;
    }
    __syncthreads();
    atomicAdd(&refcnt, myref);
    __syncthreads();

    const int maxit = *max_it;
    int T = 0;
    for (int t = 0; t < maxit; ++t) {
        // horizontal 5-tap count (zero padding)
        for (int v = 0; v < PLANE / 256; ++v) {
            int p = tid + 256 * v;
            int j = p & (WW - 1);
            float s = m[p];
            if (j >= 1)      s += m[p - 1];
            if (j >= 2)      s += m[p - 2];
            if (j <= WW - 2) s += m[p + 1];
            if (j <= WW - 3) s += m[p + 2];
            rs[p] = s;
        }
        __syncthreads();
        // vertical 5-tap count, new mask, stop statistics
        int mydiff = 0, mycov = 0;
        for (int v = 0; v < PLANE / 256; ++v) {
            int p = tid + 256 * v;
            int i = p >> 7;
            float s = rs[p];
            if (i >= 1)      s += rs[p - WW];
            if (i >= 2)      s += rs[p - 2 * WW];
            if (i <= HH - 2) s += rs[p + WW];
            if (i <= HH - 3) s += rs[p + 2 * WW];
            mc[p] = s;
            float nv = (s > 0.f) ? 1.f : 0.f;
            if (nv != m[p]) mydiff++;
            if (nv > 0.f && ref_mask[p] > 0.f) mycov++;
            m[p] = nv;                 // own-pixel only: no race
        }
        if (mydiff) atomicAdd(&ndiff, mydiff);
        if (mycov)  atomicAdd(&ncov, mycov);
        __syncthreads();
        T = t + 1;
        bool done = (ncov == refcnt) || (ndiff == 0);  // uniform
        __syncthreads();
        if (tid == 0) { ndiff = 0; ncov = 0; }
        if (done) break;
    }
    __syncthreads();

    const float mw = *mask_weight;
    for (int v = 0; v < PLANE / 256; ++v) {
        int p = tid + 256 * v;
        float s = mc[p];
        masksum[p] = (s == 0.f) ? 1.f : s * mw;
    }
    if (tid == 0) *t_out = T;
}

// ---------------------------------------------------------------------------
// Kernel 2: M = (s1 * Band5)^T applied to the identity (s1 = sqrt(fw)), so
// the full T-iteration box-filter pipeline is D = M * X * M^T.
// ---------------------------------------------------------------------------
__global__ void __launch_bounds__(256)
mbuild_kernel(const float* __restrict__ feat_weight, const int* __restrict__ t_in,
              float* __restrict__ Mout)
{
    extern __shared__ float smem[];
    float* A = smem;
    float* B = smem + PLANE;
    const int tid = threadIdx.x;

    for (int v = 0; v < PLANE / 256; ++v) {
        int p = tid + 256 * v;
        int i = p >> 7, j = p & (WW - 1);
        A[p] = (i == j) ? 1.f : 0.f;
    }
    const int   T  = *t_in;
    const float s1 = sqrtf(*feat_weight);   // fw = 1/(K*K) -> s1 = 1/K

    for (int t = 0; t < T; ++t) {
        __syncthreads();
        for (int v = 0; v < PLANE / 256; ++v) {
            int p = tid + 256 * v;
            int i = p >> 7;
            float s = A[p];
            if (i >= 1)      s += A[p - WW];
            if (i >= 2)      s += A[p - 2 * WW];
            if (i <= HH - 2) s += A[p + WW];
            if (i <= HH - 3) s += A[p + 2 * WW];
            B[p] = s * s1;
        }
        float* tmp = A; A = B; B = tmp;     // uniform swap
    }
    __syncthreads();
    for (int v = 0; v < PLANE / 256; ++v) {
        int p = tid + 256 * v;
        Mout[p] = A[p];
    }
}

// ---------------------------------------------------------------------------
// Kernel 3: per channel, D = M * (X .* mask) * M^T via V_WMMA_F32_16X16X4_F32,
// then divide by mask_sum. Two LDS planes only (M + X-overwritten-by-T) so two
// workgroups co-reside per WGP; M is staged with async global->LDS DMA.
// ---------------------------------------------------------------------------
__global__ void __launch_bounds__(256)
pconv_gemm_kernel(const float* __restrict__ inp, const float* __restrict__ maskc,
                  const float* __restrict__ Mmat, const float* __restrict__ masksum,
                  float* __restrict__ out)
{
    extern __shared__ float smem[];
    float* sM  = smem;               // [128][LDSW] shared matrix
    float* sXT = smem + HH * LDSW;   // masked input, overwritten strip-wise by T

    const int c   = blockIdx.x;
    const int tid = threadIdx.x;
    const float* xg = inp   + (size_t)c * PLANE;
    const float* mg = maskc + (size_t)c * PLANE;

    __builtin_prefetch(xg, 0, 1);      // global_prefetch_b8
    __builtin_prefetch(mg, 0, 1);

    // ---- stage M: async DMA global->LDS (ASYNCcnt path) if available ----
#if defined(__gfx1250__) && __has_builtin(__builtin_amdgcn_global_load_async_to_lds_b128)
#pragma unroll
    for (int v = 0; v < 16; ++v) {
        int q = tid + 256 * v;               // float4 index; 32 float4 per row
        int i = q >> 5;
        int j = (q & 31) * 4;
        __builtin_amdgcn_global_load_async_to_lds_b128(
            (GLOBAL_AS v4i*)(Mmat + 4 * q),
            (LDS_AS v4i*)(&sM[i * LDSW + j]),
            0, 0);
    }
#else
#pragma unroll
    for (int v = 0; v < 16; ++v) {
        int q = tid + 256 * v;
        int i = q >> 5;
        int j = (q & 31) * 4;
        *(v4f*)&sM[i * LDSW + j] = ((const v4f*)Mmat)[q];
    }
#endif

    // ---- stage X*mask (VALU product overlaps the async M transfer) ----
#pragma unroll
    for (int v = 0; v < 16; ++v) {
        int q = tid + 256 * v;
        int i = q >> 5;
        int j = (q & 31) * 4;
        v4f xx = ((const v4f*)xg)[q];
        v4f kk = ((const v4f*)mg)[q];
        *(v4f*)&sXT[i * LDSW + j] = xx * kk;
    }

#if defined(__gfx1250__) && __has_builtin(__builtin_amdgcn_global_load_async_to_lds_b128)
#if __has_builtin(__builtin_amdgcn_s_wait_asynccnt)
    __builtin_amdgcn_s_wait_asynccnt(0);
#else
    asm volatile("s_wait_asynccnt 0x0" ::: "memory");
#endif
#endif
    __syncthreads();

    const int wave  = tid >> 5;        // 8 waves, one 16-row strip each
    const int lane  = tid & 31;
    const int r     = lane & 15;       // A: M-row / B: N-col within tile
    const int half  = lane >> 4;
    const int kk2   = half * 2;        // K offset of this half-wave's VGPR pair
    const int rbase = half * 8;        // C/D row base for this half-wave
    const int i0    = wave * 16;

    const v8f vzero = {};
    v8f acc[8];

    // ---- pass 1: T = X * M^T.  Row strip of T depends only on the wave's own
    // row strip of X, so each wave overwrites its strip in place (no 3rd plane).
#pragma unroll
    for (int jt = 0; jt < 8; ++jt) acc[jt] = vzero;
    {
        const float* arow = &sXT[(i0 + r) * LDSW + kk2];
#pragma unroll 2
        for (int kt = 0; kt < 32; ++kt) {
            v2f a = *(const v2f*)(arow + 4 * kt);   // reused across 8 col tiles
#pragma unroll
            for (int jt = 0; jt < 8; ++jt) {
                v2f b = *(const v2f*)(&sM[(jt * 16 + r) * LDSW + kk2 + 4 * kt]);
                acc[jt] = __builtin_amdgcn_wmma_f32_16x16x4_f32(
                    false, a, false, b, (short)0, acc[jt], false, false);
            }
        }
    }
#pragma unroll
    for (int jt = 0; jt < 8; ++jt) {
        const int jcol = jt * 16 + r;
#pragma unroll
        for (int v = 0; v < 8; ++v)
            sXT[(i0 + rbase + v) * LDSW + jcol] = acc[jt][v];  // own rows only
    }
    __syncthreads();   // pass 2 reads all rows of T

    // ---- pass 2: D = M * T ; divide by mask_sum and store ----
#pragma unroll
    for (int jt = 0; jt < 8; ++jt) acc[jt] = vzero;
    {
        const float* arow = &sM[(i0 + r) * LDSW + kk2];
#pragma unroll 2
        for (int kt = 0; kt < 32; ++kt) {
            const int k0 = 4 * kt + kk2;
            v2f a = *(const v2f*)(arow + 4 * kt);
#pragma unroll
            for (int jt = 0; jt < 8; ++jt) {
                const float* bp = &sXT[k0 * LDSW + jt * 16 + r];
                v2f b = { bp[0], bp[LDSW] };        // ds_load_2addr_b32 pair
                acc[jt] = __builtin_amdgcn_wmma_f32_16x16x4_f32(
                    false, a, false, b, (short)0, acc[jt], false, false);
            }
        }
    }
    float* og = out + (size_t)c * PLANE;
#pragma unroll
    for (int jt = 0; jt < 8; ++jt) {
        const int jcol = jt * 16 + r;
#pragma unroll
        for (int v = 0; v < 8; ++v) {
            const int row = i0 + rbase + v;
            og[row * WW + jcol] = acc[jt][v] / masksum[row * WW + jcol];
        }
    }
}

// ---------------------------------------------------------------------------
extern "C" void kernel_launch(void* const* d_in, const int* in_sizes, int n_in,
                              void* d_out, int out_size, void* d_ws, size_t ws_size,
                              hipStream_t stream)
{
    const float* inp  = (const float*)d_in[0];   // [1,C,H,W]
    const float* msk  = (const float*)d_in[1];   // [1,C,H,W]
    const float* refm = (const float*)d_in[2];   // [H,W]
    const float* fw   = (const float*)d_in[3];   // [C,1,K,K] uniform
    const float* mw   = (const float*)d_in[4];   // [C,1,K,K] uniform
    const int*   mit  = (const int*)d_in[5];     // scalar max_iterations

    char*  ws      = (char*)d_ws;
    float* masksum = (float*)ws;                          // 64 KB
    int*   t_ws    = (int*)(ws + PLANE * 4);              // 256 B slot
    float* Mmat    = (float*)(ws + PLANE * 4 + 256);      // 64 KB

    const size_t lds_mask = 3 * PLANE * sizeof(float);        // 192 KB
    const size_t lds_mb   = 2 * PLANE * sizeof(float);        // 128 KB
    const size_t lds_gemm = 2 * HH * LDSW * sizeof(float);    // 132 KB -> 2 WG/WGP

    (void)hipFuncSetAttribute((const void*)mask_iter_kernel,
        hipFuncAttributeMaxDynamicSharedMemorySize, (int)lds_mask);
    (void)hipFuncSetAttribute((const void*)mbuild_kernel,
        hipFuncAttributeMaxDynamicSharedMemorySize, (int)lds_mb);
    (void)hipFuncSetAttribute((const void*)pconv_gemm_kernel,
        hipFuncAttributeMaxDynamicSharedMemorySize, (int)lds_gemm);

    mask_iter_kernel<<<1, 256, lds_mask, stream>>>(msk, refm, mw, mit, masksum, t_ws);
    mbuild_kernel<<<1, 256, lds_mb, stream>>>(fw, t_ws, Mmat);
    pconv_gemm_kernel<<<NCH, 256, lds_gemm, stream>>>(inp, msk, Mmat, masksum,
                                                      (float*)d_out);
}